// ASG_88038239633881
// MI455X (gfx1250) — compile-verified
//
#include <hip/hip_runtime.h>
#include <hip/hip_bf16.h>

typedef __attribute__((ext_vector_type(2))) float v2f;
typedef __attribute__((ext_vector_type(8))) float v8f;

#define BATCH 64
#define NPTS  512
#define DIM   128
#define LDW   132           // padded LDS row stride in floats (132*4B = 528B, 16B aligned, conflict-free)
#define BI    64            // block tile rows (i)
#define BJ    32            // block tile cols (j)

// ---------------------------------------------------------------------------
// Kernel 1: inverse norms.  One wave32 per row: 32 lanes x float4 = 128 elems.
// invn[b*N+n] = 1 / max(||fea_row||, 1e-8)
// ---------------------------------------------------------------------------
__global__ __launch_bounds__(256) void asg_norm_kernel(const float* __restrict__ fea,
                                                       float* __restrict__ invn) {
    int row  = blockIdx.x * 8 + (threadIdx.x >> 5);   // 8 rows per 256-thread block
    int lane = threadIdx.x & 31;
    const float4* p = (const float4*)(fea + (size_t)row * DIM);
    float4 q = p[lane];
    float s = q.x * q.x + q.y * q.y + q.z * q.z + q.w * q.w;
#pragma unroll
    for (int off = 16; off > 0; off >>= 1)
        s += __shfl_xor(s, off, 32);
    if (lane == 0)
        invn[row] = 1.0f / fmaxf(sqrtf(s), 1e-8f);
}

// ---------------------------------------------------------------------------
// Kernel 2: 64x32 output block per 128-thread workgroup (4 waves).
// Wave w computes a 16x32 strip = two 16x16 WMMA tiles, K=128 via 32 chunks
// of V_WMMA_F32_16X16X4_F32.  Epilogue: cosine scale + distance/exp masks.
// ---------------------------------------------------------------------------
__global__ __launch_bounds__(128) void asg_adj_kernel(const float* __restrict__ fea,
                                                      const float* __restrict__ coord,
                                                      const float* __restrict__ invn,
                                                      float* __restrict__ out) {
    __shared__ float As[BI][LDW];        // 64 rows  x 128 used floats
    __shared__ float Bs[BJ][LDW];        // 32 rows
    __shared__ float ns[BI + BJ];        // inv norms: [0,64)=A rows, [64,96)=B rows
    __shared__ float cxs[BI + BJ];
    __shared__ float cys[BI + BJ];

    const int tid    = threadIdx.x;
    const int b      = blockIdx.z;
    const int i_base = blockIdx.y * BI;
    const int j_base = blockIdx.x * BJ;

    // ---- cooperative loads: A rows (64x32 float4), B rows (32x32 float4) ----
    const float* fa = fea + ((size_t)b * NPTS + i_base) * DIM;
    const float* fb = fea + ((size_t)b * NPTS + j_base) * DIM;
#pragma unroll
    for (int t = 0; t < 16; ++t) {                 // 2048 float4 / 128 threads
        int idx = tid + t * 128;
        int r = idx >> 5, c4 = idx & 31;           // 32 float4 per row
        float4 v = *(const float4*)(fa + r * DIM + c4 * 4);
        *(float4*)&As[r][c4 * 4] = v;
    }
#pragma unroll
    for (int t = 0; t < 8; ++t) {                  // 1024 float4 / 128 threads
        int idx = tid + t * 128;
        int r = idx >> 5, c4 = idx & 31;
        float4 v = *(const float4*)(fb + r * DIM + c4 * 4);
        *(float4*)&Bs[r][c4 * 4] = v;
    }
    if (tid < BI + BJ) {
        int gr = (tid < BI) ? (i_base + tid) : (j_base + tid - BI);
        ns[tid] = invn[b * NPTS + gr];
        float2 cc = *(const float2*)(coord + ((size_t)b * NPTS + gr) * 2);
        cxs[tid] = cc.x;
        cys[tid] = cc.y;
    }
    __syncthreads();

    // ---- WMMA GEMM: wave w -> rows [16w, 16w+16), cols [0,32) of the block ----
    const int wave = tid >> 5;
    const int lane = tid & 31;
    const int m    = lane & 15;          // A/B row within tile (M for A-frag, N for B-frag)
    const int hi   = lane >> 4;          // upper half-wave
    const int k0   = hi * 2;             // K sub-offset for fragments

    const float* arow = &As[wave * 16 + m][0];
    const float* brow0 = &Bs[m][0];
    const float* brow1 = &Bs[16 + m][0];

    v8f acc0 = {};
    v8f acc1 = {};
#pragma unroll
    for (int kc = 0; kc < 32; ++kc) {
        int kb = kc * 4 + k0;
        v2f a  = *(const v2f*)(arow  + kb);
        v2f b0 = *(const v2f*)(brow0 + kb);
        v2f b1 = *(const v2f*)(brow1 + kb);
        acc0 = __builtin_amdgcn_wmma_f32_16x16x4_f32(false, a, false, b0, (short)0, acc0, false, false);
        acc1 = __builtin_amdgcn_wmma_f32_16x16x4_f32(false, a, false, b1, (short)0, acc1, false, false);
    }

    // ---- epilogue: C/D layout -> lane l holds col n=l%16, VGPR v -> row v+8*hi ----
    const size_t syncOff = (size_t)BATCH * NPTS * NPTS;
    v8f accs[2] = {acc0, acc1};
#pragma unroll
    for (int c = 0; c < 2; ++c) {
        const int jL = BI + c * 16 + m;
        const float nj = ns[jL];
        const float xj = cxs[jL];
        const float yj = cys[jL];
        const int gj = j_base + c * 16 + m;
#pragma unroll
        for (int v = 0; v < 8; ++v) {
            const int iL = wave * 16 + v + hi * 8;
            const int gi = i_base + iL;
            float cosv = accs[c][v] * ns[iL] * nj;
            float dx = cxs[iL] - xj;
            float dy = cys[iL] - yj;
            float d2 = dx * dx + dy * dy;
            bool diag = (gi == gj);
            float dist = sqrtf(d2 + (diag ? 1.0f : 0.0f));
            float av = diag ? 0.0f : cosv * __expf(-dist);
            float sv = (!diag && dist < 1.0f) ? 1.0f : 0.0f;
            size_t o = ((size_t)(b * NPTS + gi)) * NPTS + gj;
            out[o] = av;
            out[o + syncOff] = sv;
        }
    }
}

// ---------------------------------------------------------------------------
extern "C" void kernel_launch(void* const* d_in, const int* in_sizes, int n_in,
                              void* d_out, int out_size, void* d_ws, size_t ws_size,
                              hipStream_t stream) {
    (void)in_sizes; (void)n_in; (void)out_size; (void)ws_size;
    const float* fea   = (const float*)d_in[0];   // [64, 512, 128] f32
    const float* coord = (const float*)d_in[1];   // [64, 512, 2]   f32
    float* out  = (float*)d_out;                  // [2, 64, 512, 512] f32
    float* invn = (float*)d_ws;                   // 64*512 floats = 128 KB scratch

    // 1) inverse norms: 32768 rows, 8 rows per 256-thread block
    asg_norm_kernel<<<dim3((BATCH * NPTS) / 8), dim3(256), 0, stream>>>(fea, invn);

    // 2) adjacency: grid (j-blocks, i-blocks, batch) = (16, 8, 64)
    asg_adj_kernel<<<dim3(NPTS / BJ, NPTS / BI, BATCH), dim3(128), 0, stream>>>(fea, coord, invn, out);
}